// QuantumCircuit_360777252890
// MI455X (gfx1250) — compile-verified
//
#include <hip/hip_runtime.h>
#include <math.h>

typedef __attribute__((ext_vector_type(2))) float v2f;
typedef __attribute__((ext_vector_type(8))) float v8f;

#define NQ        22
#define DIM       (1u << NQ)        // 4,194,304 fp32 amplitudes = 16 MB (L2-resident)
#define TILE_BITS 16
#define TILE      (1u << TILE_BITS) // 65536 floats = 256 KB LDS tile per workgroup
#define WG        1024              // 32 waves per workgroup (one WGP)

// ---------------------------------------------------------------------------
// U = T3 (x) T2 (x) T1 (x) T0 entry; Tj = RY [[c,-s],[s,c]] (identity: c=1,s=0)
// ---------------------------------------------------------------------------
__device__ __forceinline__ float uEntry(unsigned m, unsigned k,
                                        const float c[4], const float s[4]) {
    float v = 1.0f;
#pragma unroll
    for (int j = 0; j < 4; ++j) {
        unsigned mj = (m >> j) & 1u;
        unsigned kj = (k >> j) & 1u;
        v *= (mj == kj) ? c[j] : (kj ? -s[j] : s[j]);
    }
    return v;
}

// 4-way value select (lowered to 3 v_cndmask)
__device__ __forceinline__ float pick4(unsigned d, float s0, float s1,
                                       float s2, float s3) {
    const float lo = (d & 1u) ? s1 : s0;
    const float hi = (d & 1u) ? s3 : s2;
    return (d & 2u) ? hi : lo;
}

// ---------------------------------------------------------------------------
// Apply one fused-4-qubit rotation group inside a 2^16-float LDS tile.
//   gl: group bit offset in LDS-local index; gbase: offset in GLOBAL index
//   mask: which of the 4 slots carry a real RY (others identity)
// One wave = one 16x16 sub-tile: D = U * S via 4 chained v_wmma_f32_16x16x4_f32.
// ---------------------------------------------------------------------------
__device__ __forceinline__ void applyGroupLDS(float* tile, const float* phi,
                                              int qb, int gl, int gbase, int mask) {
    const unsigned lane = threadIdx.x & 31u;
    const unsigned wv   = threadIdx.x >> 5;
    const unsigned n    = lane & 15u;
    const unsigned half = lane >> 4;

    float c[4], s[4];
#pragma unroll
    for (int j = 0; j < 4; ++j) {
        if (mask & (1 << j)) {
            int wire = 21 - (gbase + j);          // wire 0 = MSB convention
            __sincosf(0.5f * phi[qb * NQ + wire], &s[j], &c[j]);
        } else { c[j] = 1.0f; s[j] = 0.0f; }
    }

    v2f a[4];                                     // iteration-invariant U rows
#pragma unroll
    for (int kc = 0; kc < 4; ++kc) {
        const unsigned k0 = 4u * kc + 2u * half;
        a[kc][0] = uEntry(n, k0,      c, s);
        a[kc][1] = uEntry(n, k0 + 1u, c, s);
    }

    const unsigned loM = (1u << gl) - 1u;
    for (int it = 0; it < 8; ++it) {
        const unsigned o    = ((unsigned)it * 32u + wv) * 16u + n;   // [0,4096)
        const unsigned base = ((o >> gl) << (gl + 4)) | (o & loM);
        v8f acc = {0.f,0.f,0.f,0.f,0.f,0.f,0.f,0.f};
#pragma unroll
        for (int kc = 0; kc < 4; ++kc) {
            const unsigned k0 = 4u * kc + 2u * half;
            v2f bm;
            bm[0] = tile[base | (k0 << gl)];
            bm[1] = tile[base | ((k0 + 1u) << gl)];
            acc = __builtin_amdgcn_wmma_f32_16x16x4_f32(
                false, a[kc], false, bm, (short)0, acc, false, false);
        }
#pragma unroll
        for (int v = 0; v < 8; ++v)
            tile[base | ((unsigned)(v + 8 * half) << gl)] = acc[v];
    }
}

// ---------------------------------------------------------------------------
// Low pass: contiguous 2^16 tile, groups at global bits {0,4,8,12}.
// inMode 1: Gray-gather (fused CNOT chain) as b128 loads + component swizzle:
//           gray(4k + d) = G ^ {0,1,3,2} always stays in G's aligned 4-block.
// inMode 2: analytic block-0 product state amp0(gray(j)) via LDS tables.
// ---------------------------------------------------------------------------
__global__ __launch_bounds__(WG) void qsim_low(const float* in, float* out,
                                               const float* phi, int qb, int inMode) {
    extern __shared__ float smem[];
    float* tile = smem;               // 65536 floats
    float* TL   = smem + TILE;        // 2048 (analytic mode only)
    float* TH   = TL + 2048;          // 2048
    const unsigned t        = threadIdx.x;
    const unsigned tileBase = blockIdx.x << TILE_BITS;

    if (inMode == 2) {
        float cw[NQ], sw[NQ];
#pragma unroll
        for (int p = 0; p < NQ; ++p)
            __sincosf(0.5f * phi[21 - p], &sw[p], &cw[p]);   // phi row 0
        for (unsigned v = t; v < 2048u; v += WG) {
            float pl = 1.f, ph = 1.f;
#pragma unroll
            for (int p = 0; p < 11; ++p) {
                pl *= ((v >> p) & 1u) ? sw[p]      : cw[p];
                ph *= ((v >> p) & 1u) ? sw[p + 11] : cw[p + 11];
            }
            TL[v] = pl; TH[v] = ph;
        }
        __syncthreads();
        for (unsigned e = 0; e < TILE / (4u * WG); ++e) {
            const unsigned lam = (e * WG + t) * 4u;
            const unsigned j   = tileBase | lam;                 // 4-aligned
            const unsigned G   = j ^ (j >> 1);                   // Gray map
            const float    th  = TH[G >> 11];                    // shared by block
            const unsigned b   = G & 2047u;
            float4 tl;
            tl.x = TL[b]      * th;
            tl.y = TL[b ^ 1u] * th;
            tl.z = TL[b ^ 3u] * th;
            tl.w = TL[b ^ 2u] * th;
            *reinterpret_cast<float4*>(tile + lam) = tl;
        }
    } else {
        for (unsigned e = 0; e < TILE / (4u * WG); ++e) {
            const unsigned lam = (e * WG + t) * 4u;
            const unsigned j   = tileBase | lam;                 // 4-aligned
            const unsigned G   = j ^ (j >> 1);                   // G has bit0==0
            const float4   f   = *reinterpret_cast<const float4*>(in + (G & ~3u));
            const bool     hi  = (G & 2u) != 0u;
            float4 tl;                                           // un-Gray swizzle
            tl.x = hi ? f.z : f.x;
            tl.y = hi ? f.w : f.y;
            tl.z = hi ? f.y : f.w;
            tl.w = hi ? f.x : f.z;
            *reinterpret_cast<float4*>(tile + lam) = tl;
        }
    }
    __syncthreads();
    applyGroupLDS(tile, phi, qb,  0,  0, 0xF); __syncthreads();
    applyGroupLDS(tile, phi, qb,  4,  4, 0xF); __syncthreads();
    applyGroupLDS(tile, phi, qb,  8,  8, 0xF); __syncthreads();
    applyGroupLDS(tile, phi, qb, 12, 12, 0xF); __syncthreads();
    for (unsigned e = 0; e < TILE / (4u * WG); ++e) {
        const unsigned lam = (e * WG + t) * 4u;
        *reinterpret_cast<float4*>(out + (tileBase | lam)) =
            *reinterpret_cast<const float4*>(tile + lam);
    }
}

// ---------------------------------------------------------------------------
// High pass: gathered tile (bits 16..21 free x 1024 contiguous low elements),
// groups at global bits {16..19} and {20,21}. finalMode squares amplitudes and
// scatters through the inverse-Gray map as packed b128 stores: the 4 values of
// an aligned i-block fill exactly the aligned output block at invgray(i)&~3.
// ---------------------------------------------------------------------------
__global__ __launch_bounds__(WG) void qsim_high(const float* in, float* out,
                                                const float* phi, int qb, int finalMode) {
    extern __shared__ float smem[];
    float* tile = smem;
    const unsigned t   = threadIdx.x;
    const unsigned mid = blockIdx.x;            // fixed bits 10..15
    for (unsigned e = 0; e < TILE / (4u * WG); ++e) {
        const unsigned lam = (e * WG + t) * 4u; // lam = h*1024 + l, l 4-aligned
        const unsigned i = ((lam >> 10) << 16) | (mid << 10) | (lam & 1023u);
        *reinterpret_cast<float4*>(tile + lam) =
            *reinterpret_cast<const float4*>(in + i);
    }
    __syncthreads();
    applyGroupLDS(tile, phi, qb, 10, 16, 0xF); __syncthreads();
    applyGroupLDS(tile, phi, qb, 12, 18, 0xC); __syncthreads();  // wires 1,0
    if (finalMode) {
        for (unsigned e = 0; e < TILE / (4u * WG); ++e) {
            const unsigned lam = (e * WG + t) * 4u;
            const unsigned i = ((lam >> 10) << 16) | (mid << 10) | (lam & 1023u);
            unsigned J = i;                      // inverse Gray (prefix XOR)
            J ^= J >> 1; J ^= J >> 2; J ^= J >> 4; J ^= J >> 8; J ^= J >> 16;
            const float4 v = *reinterpret_cast<const float4*>(tile + lam);
            const float s0 = v.x * v.x, s1 = v.y * v.y;
            const float s2 = v.z * v.z, s3 = v.w * v.w;
            const unsigned r = J & 3u;
            // component p receives element d = gray2(r^p), gray2(x)=x^(x>>1)
            float4 o;
            { unsigned x = r;       o.x = pick4((x ^ (x >> 1)) & 3u, s0, s1, s2, s3); }
            { unsigned x = r ^ 1u;  o.y = pick4((x ^ (x >> 1)) & 3u, s0, s1, s2, s3); }
            { unsigned x = r ^ 2u;  o.z = pick4((x ^ (x >> 1)) & 3u, s0, s1, s2, s3); }
            { unsigned x = r ^ 3u;  o.w = pick4((x ^ (x >> 1)) & 3u, s0, s1, s2, s3); }
            *reinterpret_cast<float4*>(out + (J & ~3u)) = o;     // probs
        }
    } else {
        for (unsigned e = 0; e < TILE / (4u * WG); ++e) {
            const unsigned lam = (e * WG + t) * 4u;
            const unsigned i = ((lam >> 10) << 16) | (mid << 10) | (lam & 1023u);
            *reinterpret_cast<float4*>(out + i) =
                *reinterpret_cast<const float4*>(tile + lam);
        }
    }
}

// ---------------------------------------------------------------------------
// 6 kernels total (needs 16 MB of d_ws scratch):
//   K1 low (b1, analytic block-0 input) -> ws      K2 high(b1) in-place ws
//   K3 low (b2, gray) ws -> out                    K4 high(b2) in-place out
//   K5 low (b3, gray) out -> ws                    K6 high(b3, final) ws -> out
// ---------------------------------------------------------------------------
extern "C" void kernel_launch(void* const* d_in, const int* in_sizes, int n_in,
                              void* d_out, int out_size, void* d_ws, size_t ws_size,
                              hipStream_t stream) {
    (void)in_sizes; (void)n_in; (void)out_size; (void)ws_size;
    const float* phi = (const float*)d_in[0];   // (4, 22) fp32
    float* bufW = (float*)d_ws;                 // 16 MB state scratch
    float* bufO = (float*)d_out;

    const size_t smemLow  = (TILE + 4096) * sizeof(float);  // tile + tables
    const size_t smemHigh = TILE * sizeof(float);
    const unsigned nWG = DIM / TILE;                        // 64 tiles

    qsim_low <<<nWG, WG, smemLow,  stream>>>(bufO, bufW, phi, 1, 2);
    qsim_high<<<nWG, WG, smemHigh, stream>>>(bufW, bufW, phi, 1, 0);
    qsim_low <<<nWG, WG, smemLow,  stream>>>(bufW, bufO, phi, 2, 1);
    qsim_high<<<nWG, WG, smemHigh, stream>>>(bufO, bufO, phi, 2, 0);
    qsim_low <<<nWG, WG, smemLow,  stream>>>(bufO, bufW, phi, 3, 1);
    qsim_high<<<nWG, WG, smemHigh, stream>>>(bufW, bufO, phi, 3, 1);
}